// RoutingLayer_63282048139562
// MI455X (gfx1250) — compile-verified
//
#include <hip/hip_runtime.h>
#include <math.h>

// Problem constants (from reference)
#define C_ 10
#define B_ 128
#define R_ 1152
#define I_ 16
#define O_ 32

#define RC_ 32               // routes per block in k_s
#define RCHUNKS_ (R_ / RC_)  // 36

typedef __attribute__((ext_vector_type(2))) float v2f;
typedef __attribute__((ext_vector_type(8))) float v8f;

// ---------------------------------------------------------------------------
// zero-fill helper (graph-capture safe; avoids host memset APIs)
// ---------------------------------------------------------------------------
__global__ __launch_bounds__(256) void k_zero(float* __restrict__ p, int n) {
    int i = blockIdx.x * 256 + threadIdx.x;
    if (i < n) p[i] = 0.0f;
}

// ---------------------------------------------------------------------------
// k_s : s[c,b,o] = sum_r cc[c,b,r] * (x[b,r,:] . w[c,r,:,o])
// grid = (C_, RCHUNKS_), block = 256 (8 waves). Wave w handles batch tile
// rows [16w, 16w+16). Each wave: 2 f32 accumulators (o 0-15 / 16-31),
// 8 v_wmma_f32_16x16x4_f32 per route, atomicAdd partials to s.
// HAS_CC=false -> uniform 1/R (iteration 0: softmax of zero logits).
// ---------------------------------------------------------------------------
template <bool HAS_CC>
__global__ __launch_bounds__(256) void k_s(const float* __restrict__ x,
                                           const float* __restrict__ w,
                                           const float* __restrict__ cc,
                                           float* __restrict__ s) {
    const int c    = blockIdx.x;
    const int rc   = blockIdx.y;
    const int wave = threadIdx.x >> 5;   // m-tile 0..7
    const int lane = threadIdx.x & 31;
    const int col  = lane & 15;
    const int h    = lane >> 4;          // K-half select per f32 WMMA layout
    const int brow = wave * 16 + col;    // A-operand row (batch index)

    v8f acc0 = {};
    v8f acc1 = {};
    const float invR = 1.0f / (float)R_;
    const int r0 = rc * RC_;
    const float* ccp =
        HAS_CC ? (cc + ((size_t)(c * B_ + brow)) * R_ + r0) : nullptr;

    for (int rr = 0; rr < RC_; ++rr) {
        const int r = r0 + rr;
        const float scale = HAS_CC ? ccp[rr] : invR;
        const float* xp = x + ((size_t)brow * R_ + r) * I_;
        const float* wp = w + (((size_t)c * R_ + r) * I_) * O_;
#pragma unroll
        for (int ks = 0; ks < 4; ++ks) {
            const int i0 = 4 * ks + 2 * h;   // lanes 0-15: K{0,1}; 16-31: K{2,3}
            v2f a;  a.x  = xp[i0] * scale;         a.y  = xp[i0 + 1] * scale;
            v2f b0; b0.x = wp[i0 * O_ + col];      b0.y = wp[(i0 + 1) * O_ + col];
            v2f b1; b1.x = wp[i0 * O_ + 16 + col]; b1.y = wp[(i0 + 1) * O_ + 16 + col];
            acc0 = __builtin_amdgcn_wmma_f32_16x16x4_f32(false, a, false, b0,
                                                         (short)0, acc0, false, false);
            acc1 = __builtin_amdgcn_wmma_f32_16x16x4_f32(false, a, false, b1,
                                                         (short)0, acc1, false, false);
        }
    }
    // C/D layout: VGPR j -> M=j (lanes 0-15) / M=j+8 (lanes 16-31), N = col
#pragma unroll
    for (int j = 0; j < 8; ++j) {
        const int bo = wave * 16 + j + 8 * h;
        float* sp = s + ((size_t)(c * B_ + bo)) * O_;
        atomicAdd(sp + col, acc0[j]);
        atomicAdd(sp + 16 + col, acc1[j]);
    }
}

// ---------------------------------------------------------------------------
// k_squash : reference's _squash with dim=1 (batch axis!).
// n2[c,o] = sum_b s[c,b,o]^2 ; out = s * n2/((1+n2)*sqrt(n2))
// grid = C_, block = 256
// ---------------------------------------------------------------------------
__global__ __launch_bounds__(256) void k_squash(const float* __restrict__ s,
                                                float* __restrict__ outv) {
    const int c = blockIdx.x;
    const int o = threadIdx.x & 31;
    const int g = threadIdx.x >> 5;      // 8 batch slices of 16
    __shared__ float red[256];
    __shared__ float fac[O_];

    float part = 0.0f;
    for (int b = g * 16; b < g * 16 + 16; ++b) {
        float v = s[((size_t)(c * B_ + b)) * O_ + o];
        part += v * v;
    }
    red[threadIdx.x] = part;
    __syncthreads();
    if (g == 0) {
        float n2 = 0.0f;
#pragma unroll
        for (int gg = 0; gg < 8; ++gg) n2 += red[o + 32 * gg];
        fac[o] = (n2 / (1.0f + n2)) * rsqrtf(n2);
    }
    __syncthreads();
    for (int idx = threadIdx.x; idx < B_ * O_; idx += 256) {
        const int b = idx >> 5, oo = idx & 31;
        const size_t k = ((size_t)(c * B_ + b)) * O_ + oo;
        outv[k] = s[k] * fac[oo];
    }
}

// ---------------------------------------------------------------------------
// k_logit : logits[c,b,r] (+)= sum_o pred[c,b,r,o] * out[c,b,o]
// One wave per (c, m-tile, r): recompute pred 16x32 tile via 8 WMMAs, then
// per-row dot with out via shfl_xor butterfly over the 16-lane N axis.
// grid = C_*R_ blocks of 256 (8 waves)
// ---------------------------------------------------------------------------
__global__ __launch_bounds__(256) void k_logit(const float* __restrict__ x,
                                               const float* __restrict__ w,
                                               const float* __restrict__ outv,
                                               float* __restrict__ logits,
                                               int accumulate) {
    int wid = blockIdx.x * 8 + (threadIdx.x >> 5);
    const int lane = threadIdx.x & 31;
    const int r  = wid % R_; wid /= R_;
    const int mt = wid % 8;
    const int c  = wid / 8;
    const int col = lane & 15;
    const int h   = lane >> 4;
    const int brow = mt * 16 + col;

    v8f acc0 = {};
    v8f acc1 = {};
    const float* xp = x + ((size_t)brow * R_ + r) * I_;
    const float* wp = w + (((size_t)c * R_ + r) * I_) * O_;
#pragma unroll
    for (int ks = 0; ks < 4; ++ks) {
        const int i0 = 4 * ks + 2 * h;
        v2f a;  a.x  = xp[i0];                 a.y  = xp[i0 + 1];
        v2f b0; b0.x = wp[i0 * O_ + col];      b0.y = wp[(i0 + 1) * O_ + col];
        v2f b1; b1.x = wp[i0 * O_ + 16 + col]; b1.y = wp[(i0 + 1) * O_ + 16 + col];
        acc0 = __builtin_amdgcn_wmma_f32_16x16x4_f32(false, a, false, b0,
                                                     (short)0, acc0, false, false);
        acc1 = __builtin_amdgcn_wmma_f32_16x16x4_f32(false, a, false, b1,
                                                     (short)0, acc1, false, false);
    }

    float p[8];
#pragma unroll
    for (int j = 0; j < 8; ++j) {
        const int row = mt * 16 + j + 8 * h;
        const float* op = outv + ((size_t)(c * B_ + row)) * O_;
        p[j] = acc0[j] * op[col] + acc1[j] * op[16 + col];
    }
    // reduce over the 16 N-lanes inside each half-wave
#pragma unroll
    for (int j = 0; j < 8; ++j) {
        p[j] += __shfl_xor(p[j], 1, 32);
        p[j] += __shfl_xor(p[j], 2, 32);
        p[j] += __shfl_xor(p[j], 4, 32);
        p[j] += __shfl_xor(p[j], 8, 32);
    }
    if (col < 8) {
        float v = p[0];
        v = (col == 1) ? p[1] : v;
        v = (col == 2) ? p[2] : v;
        v = (col == 3) ? p[3] : v;
        v = (col == 4) ? p[4] : v;
        v = (col == 5) ? p[5] : v;
        v = (col == 6) ? p[6] : v;
        v = (col == 7) ? p[7] : v;
        const int row = mt * 16 + col + 8 * h;
        const size_t idx = ((size_t)(c * B_ + row)) * R_ + r;
        logits[idx] = accumulate ? (logits[idx] + v) : v;
    }
}

// ---------------------------------------------------------------------------
// k_softmax : softmax over routes (length R_) per (c,b) row. grid = C_*B_
// ---------------------------------------------------------------------------
__global__ __launch_bounds__(256) void k_softmax(const float* __restrict__ logits,
                                                 float* __restrict__ cc) {
    const int row = blockIdx.x;
    const float* lp = logits + (size_t)row * R_;
    float* cp = cc + (size_t)row * R_;
    __shared__ float red[256];

    float m = -INFINITY;
    for (int i = threadIdx.x; i < R_; i += 256) m = fmaxf(m, lp[i]);
    red[threadIdx.x] = m;
    __syncthreads();
    for (int st = 128; st > 0; st >>= 1) {
        if (threadIdx.x < st) red[threadIdx.x] = fmaxf(red[threadIdx.x], red[threadIdx.x + st]);
        __syncthreads();
    }
    m = red[0];
    __syncthreads();

    float sum = 0.0f;
    for (int i = threadIdx.x; i < R_; i += 256) {
        float e = __expf(lp[i] - m);
        cp[i] = e;
        sum += e;
    }
    red[threadIdx.x] = sum;
    __syncthreads();
    for (int st = 128; st > 0; st >>= 1) {
        if (threadIdx.x < st) red[threadIdx.x] += red[threadIdx.x + st];
        __syncthreads();
    }
    const float inv = 1.0f / red[0];
    for (int i = threadIdx.x; i < R_; i += 256) cp[i] *= inv;
}

// ---------------------------------------------------------------------------
extern "C" void kernel_launch(void* const* d_in, const int* in_sizes, int n_in,
                              void* d_out, int out_size, void* d_ws, size_t ws_size,
                              hipStream_t stream) {
    const float* x = (const float*)d_in[0];           // [B,R,I]
    const float* w = (const float*)d_in[1];           // [C,R,I,O]
    float* outp = (float*)d_out;                      // [C,B,1,1,O] = [C,B,O]

    const size_t CBR = (size_t)C_ * B_ * R_;          // 1,474,560
    const size_t CBO = (size_t)C_ * B_ * O_;          // 40,960

    float* logits = (float*)d_ws;                     // [C,B,R]
    float* cc     = logits + CBR;                     // [C,B,R]
    float* s      = cc + CBR;                         // [C,B,O]
    float* ob     = s + CBO;                          // [C,B,O] current outputs

    const dim3 gS(C_, RCHUNKS_, 1);
    const int  nzb = (int)((CBO + 255) / 256);
    const int  gLogit = C_ * R_;                      // 8 waves/block covers (c,mt,r)

    // ---- iteration 0 (uniform routing coefficients 1/R) ----
    k_zero        <<<nzb, 256, 0, stream>>>(s, (int)CBO);
    k_s<false>    <<<gS, 256, 0, stream>>>(x, w, nullptr, s);
    k_squash      <<<C_, 256, 0, stream>>>(s, ob);
    k_logit       <<<gLogit, 256, 0, stream>>>(x, w, ob, logits, 0);
    k_softmax     <<<C_ * B_, 256, 0, stream>>>(logits, cc);

    // ---- iteration 1 ----
    k_zero        <<<nzb, 256, 0, stream>>>(s, (int)CBO);
    k_s<true>     <<<gS, 256, 0, stream>>>(x, w, cc, s);
    k_squash      <<<C_, 256, 0, stream>>>(s, ob);
    k_logit       <<<gLogit, 256, 0, stream>>>(x, w, ob, logits, 1);
    k_softmax     <<<C_ * B_, 256, 0, stream>>>(logits, cc);

    // ---- iteration 2 (final; write d_out) ----
    k_zero        <<<nzb, 256, 0, stream>>>(s, (int)CBO);
    k_s<true>     <<<gS, 256, 0, stream>>>(x, w, cc, s);
    k_squash      <<<C_, 256, 0, stream>>>(s, outp);
}